// RayCaster_5987184411372
// MI455X (gfx1250) — compile-verified
//
#include <hip/hip_runtime.h>

// ---------------- problem constants (from reference) ----------------
#define MAPW        800
#define WPR         25                  // 800 bits / 32 = 25 words per row
#define NWORDS      (MAPW * WPR)        // 20000 u32 = 80000 bytes bitmap
#define NPART       131072
#define NBEAM       90
#define NSTEPS      151
#define MAXRPX      150.0f
#define OCC_THR     0.35f
#define DEG2RAD     0.017453292519943295f

// --------------------------------------------------------------------
// Kernel 1: pack occupancy map into a 1-bit bitmap using wave32 ballot.
// Lane l of each wave tests map[tid]>thr; ballot_w32 yields exactly the
// 32-bit word (bit position == x & 31 because rows are 32-aligned: 800%32==0).
// --------------------------------------------------------------------
__global__ __launch_bounds__(256) void pack_bits(const float* __restrict__ map,
                                                 unsigned* __restrict__ bits)
{
    int tid = blockIdx.x * 256 + threadIdx.x;          // 0 .. 639999
    float v = map[tid];
    unsigned m = __builtin_amdgcn_ballot_w32(v > OCC_THR);
    if ((threadIdx.x & 31) == 0)
        bits[tid >> 5] = m;
}

// --------------------------------------------------------------------
// Kernel 2: ray cast. One thread per particle, 90 beams each.
// The 80 KB bitmap is DMA'd global->LDS with CDNA5 async-to-LDS ops
// (tracked by ASYNCcnt), then every map probe is one ds_load_b32.
// --------------------------------------------------------------------
__global__ __launch_bounds__(256) void raycast(const float* __restrict__ X,
                                               const unsigned* __restrict__ bits,
                                               float* __restrict__ out)
{
    extern __shared__ unsigned smem[];                 // [NWORDS] bitmap + beam tables
    float* ctab = (float*)&smem[NWORDS];               // cos(delta_b), 90 entries
    float* stab = ctab + NBEAM;                        // sin(delta_b), 90 entries

    // ---- async bulk copy of the bitmap into LDS (gfx1250 ASYNC path) ----
    {
        unsigned ldsBase = (unsigned)(size_t)&smem[0]; // low 32 bits of a generic
                                                       // LDS pointer == LDS byte addr
        const unsigned char* g = (const unsigned char*)bits;
        for (int off = (int)threadIdx.x * 16; off < NWORDS * 4; off += 256 * 16) {
            unsigned ldsAddr = ldsBase + (unsigned)off;
            const unsigned char* gaddr = g + off;
            asm volatile("global_load_async_to_lds_b128 %0, %1, off"
                         :: "v"(ldsAddr), "v"(gaddr)
                         : "memory");
        }
        asm volatile("s_wait_asynccnt 0x0" ::: "memory");
    }

    // ---- per-beam direction table (uniform across block) ----
    if (threadIdx.x < NBEAM) {
        float db = (float)(-90 + 2 * (int)threadIdx.x) * DEG2RAD;
        float s, c;
        sincosf(db, &s, &c);
        ctab[threadIdx.x] = c;
        stab[threadIdx.x] = s;
    }
    __syncthreads();

    // ---- per-particle setup ----
    int p = blockIdx.x * 256 + threadIdx.x;            // 0 .. 131071
    float x   = X[3 * p + 0];
    float y   = X[3 * p + 1];
    float yaw = X[3 * p + 2];
    float sy, cy;
    sincosf(yaw, &sy, &cy);
    float xl = fmaf(25.0f, cy, x) * 0.1f;              // (x + 25*cos(yaw)) / 10
    float yl = fmaf(25.0f, sy, y) * 0.1f;

    float* o = out + (size_t)p * NBEAM;

    // ---- 90 beams, early-exit ray march ----
    for (int b = 0; b < NBEAM; ++b) {
        float cd = ctab[b];
        float sd = stab[b];
        // cos(db + yaw), sin(db + yaw) via angle addition
        float ca = cd * cy - sd * sy;
        float sa = sd * cy + cd * sy;

        float hit = MAXRPX;
        for (int r = 0; r < NSTEPS; ++r) {
            float fr = (float)r;
            float fx = rintf(fmaf(ca, fr, xl));        // round-half-even == jnp.round
            float fy = rintf(fmaf(sa, fr, yl));
            fx = fminf(fmaxf(fx, 0.0f), 799.0f);       // clip before int cast
            fy = fminf(fmaxf(fy, 0.0f), 799.0f);
            int xp = (int)fx;
            int yp = (int)fy;
            unsigned w = smem[yp * WPR + (xp >> 5)];   // ds_load_b32
            if ((w >> (xp & 31)) & 1u) { hit = fr; break; }
        }
        o[b] = hit;
    }
}

// --------------------------------------------------------------------
// Harness entry point.
//   d_in[0] = X_t1           (131072 x 3 f32)
//   d_in[1] = occupancy_map  (800 x 800 f32)
//   d_out   = hit            (131072 x 90 f32)
//   d_ws    = scratch, we use first 80000 bytes for the bitmap
// --------------------------------------------------------------------
extern "C" void kernel_launch(void* const* d_in, const int* in_sizes, int n_in,
                              void* d_out, int out_size, void* d_ws, size_t ws_size,
                              hipStream_t stream)
{
    const float* X   = (const float*)d_in[0];
    const float* map = (const float*)d_in[1];
    float*    out    = (float*)d_out;
    unsigned* bits   = (unsigned*)d_ws;

    // 1) pack occupancy -> 1-bit bitmap (fully rewritten every call: deterministic)
    pack_bits<<<(MAPW * MAPW) / 256, 256, 0, stream>>>(map, bits);

    // 2) ray cast with LDS-resident bitmap
    size_t shbytes = (size_t)NWORDS * 4 + (size_t)2 * NBEAM * 4;   // 80720 B
    raycast<<<NPART / 256, 256, shbytes, stream>>>(X, bits, out);
}